// GATLayer_58480274702575
// MI455X (gfx1250) — compile-verified
//
#include <hip/hip_runtime.h>

typedef __attribute__((ext_vector_type(2))) float v2f;
typedef __attribute__((ext_vector_type(4))) float v4f;
typedef __attribute__((ext_vector_type(8))) float v8f;

#define N_NODES 4096
#define IN_FEAT 256
#define OUT_FEAT 64
#define N_HEADS 8
#define HID (N_HEADS * OUT_FEAT) /* 512 */
#define NEG_SLOPE 0.2f
#define JCHUNK 128               /* adjacency columns staged in LDS per step */
#define APITCH 132               /* padded LDS row pitch (floats): bank-conflict-free */

// ---------------------------------------------------------------------------
// Kernel 1: Wh = x @ W   (4096x256 @ 256x512), fp32 WMMA 16x16x4
// Block = 128 threads = 4 waves; each wave owns one 16x16 output tile.
// ---------------------------------------------------------------------------
__global__ void gat_gemm_xw(const float* __restrict__ x,
                            const float* __restrict__ W,
                            float* __restrict__ Wh) {
    const int lane = threadIdx.x & 31;
    const int wave = threadIdx.x >> 5;
    const int m0 = blockIdx.y * 16;
    const int n0 = blockIdx.x * 64 + wave * 16;
    const int m  = lane & 15;          // A row / B col within tile
    const int kb = (lane >> 4) * 2;    // K sub-offset for this half-wave

    v8f acc = {};
    for (int k = 0; k < IN_FEAT; k += 4) {
        v2f a, b;
        // A: 16x4 fp32 tile of x
        a.x = x[(size_t)(m0 + m) * IN_FEAT + k + kb];
        a.y = x[(size_t)(m0 + m) * IN_FEAT + k + kb + 1];
        // B: 4x16 fp32 tile of W
        b.x = W[(size_t)(k + kb) * HID + n0 + m];
        b.y = W[(size_t)(k + kb + 1) * HID + n0 + m];
        acc = __builtin_amdgcn_wmma_f32_16x16x4_f32(false, a, false, b,
                                                    (short)0, acc, false, false);
    }
    // C/D layout: VGPR r holds M = r (lanes 0-15) / M = r+8 (lanes 16-31)
    const int row_hi = (lane >> 4) * 8;
#pragma unroll
    for (int r = 0; r < 8; ++r)
        Wh[(size_t)(m0 + row_hi + r) * HID + n0 + m] = acc[r];
}

// ---------------------------------------------------------------------------
// Kernel 2: attention logit halves  e_src[n,h], e_dst[n,h]
// ---------------------------------------------------------------------------
__global__ void gat_scores(const float* __restrict__ Wh,
                           const float* __restrict__ a,
                           float* __restrict__ e_src,
                           float* __restrict__ e_dst) {
    int idx = blockIdx.x * blockDim.x + threadIdx.x;
    if (idx >= N_NODES * N_HEADS) return;
    int node = idx >> 3;
    int h    = idx & 7;
    const float* wh = Wh + (size_t)node * HID + h * OUT_FEAT;
    const float* as = a + h * 2 * OUT_FEAT;
    const float* ad = as + OUT_FEAT;
    float s1 = 0.f, s2 = 0.f;
#pragma unroll 4
    for (int d = 0; d < OUT_FEAT; ++d) {
        float v = wh[d];
        s1 += v * as[d];
        s2 += v * ad[d];
    }
    e_src[idx] = s1;
    e_dst[idx] = s2;
}

// ---------------------------------------------------------------------------
// Kernel 3: per-row masked softmax stats (max, 1/sum) for all 8 heads.
// One block per row; online softmax per thread, LDS tree reduction.
// ---------------------------------------------------------------------------
__global__ void gat_stats(const float* __restrict__ adj,
                          const float* __restrict__ e_src,
                          const float* __restrict__ e_dst,
                          float* __restrict__ row_m,
                          float* __restrict__ row_inv) {
    __shared__ float sm[256][N_HEADS];
    __shared__ float ss[256][N_HEADS];
    const int i   = blockIdx.x;
    const int tid = threadIdx.x;

    float es[N_HEADS];
#pragma unroll
    for (int h = 0; h < N_HEADS; ++h) es[h] = e_src[i * N_HEADS + h];

    float m[N_HEADS], s[N_HEADS];
#pragma unroll
    for (int h = 0; h < N_HEADS; ++h) { m[h] = -1e30f; s[h] = 0.f; }

    const size_t arow = (size_t)i * N_NODES;
    for (int j = tid; j < N_NODES; j += 256) {
        if (adj[arow + j] > 0.f) {
#pragma unroll
            for (int h = 0; h < N_HEADS; ++h) {
                float f = es[h] + e_dst[j * N_HEADS + h];
                f = (f >= 0.f) ? f : NEG_SLOPE * f;
                if (f > m[h]) {
                    s[h] = s[h] * __expf(m[h] - f) + 1.f;
                    m[h] = f;
                } else {
                    s[h] += __expf(f - m[h]);
                }
            }
        }
    }
#pragma unroll
    for (int h = 0; h < N_HEADS; ++h) { sm[tid][h] = m[h]; ss[tid][h] = s[h]; }
    __syncthreads();
    for (int off = 128; off > 0; off >>= 1) {
        if (tid < off) {
#pragma unroll
            for (int h = 0; h < N_HEADS; ++h) {
                float m1 = sm[tid][h],       s1 = ss[tid][h];
                float m2 = sm[tid + off][h], s2 = ss[tid + off][h];
                float mm = fmaxf(m1, m2);
                sm[tid][h] = mm;
                ss[tid][h] = s1 * __expf(m1 - mm) + s2 * __expf(m2 - mm);
            }
        }
        __syncthreads();
    }
    if (tid < N_HEADS) {
        row_m[i * N_HEADS + tid]   = sm[0][tid];
        row_inv[i * N_HEADS + tid] = 1.f / ss[0][tid];
    }
}

// ---------------------------------------------------------------------------
// Kernel 4: out = relu(alpha @ Wh) per head; alpha recomputed on the fly.
// Block = 256 threads = 8 waves; wave w handles head w for a 16-row tile.
// adj is staged through LDS once per block (16x128 fp32 tile, coalesced
// b128 loads, padded pitch => conflict-free ds_load_b64 A-fragment reads),
// removing the 8x per-head redundancy and the scattered 4B global loads.
// All-zero alpha chunks (wave-uniform test) skip B loads + WMMAs.
// ---------------------------------------------------------------------------
__global__ void gat_aggregate(const float* __restrict__ adj,
                              const float* __restrict__ Wh,
                              const float* __restrict__ e_src,
                              const float* __restrict__ e_dst,
                              const float* __restrict__ row_m,
                              const float* __restrict__ row_inv,
                              float* __restrict__ out) {
    __shared__ float sadj[16][APITCH];   // 16 rows x 128 cols (padded)

    const int tid  = threadIdx.x;
    const int lane = tid & 31;
    const int h    = tid >> 5;           // head = wave id (8 waves)
    const int i0   = blockIdx.x * 16;
    const int m    = lane & 15;          // A row / B col within tile
    const int kb   = (lane >> 4) * 2;    // K sub-offset

    const int   i    = i0 + m;
    const float esrc = e_src[i * N_HEADS + h];
    const float mi   = row_m[i * N_HEADS + h];
    const float inv  = row_inv[i * N_HEADS + h];

    // Staging map: thread t copies 8 consecutive floats of row (t>>4).
    const int srow = tid >> 4;
    const int scol = (tid & 15) * 8;
    const size_t sgl = (size_t)(i0 + srow) * N_NODES + scol;

    v8f acc0 = {}, acc1 = {}, acc2 = {}, acc3 = {};

    for (int J = 0; J < N_NODES; J += JCHUNK) {
        // ---- cooperative stage of adj[i0..i0+15][J..J+127] into LDS ----
        __syncthreads();
        const v4f* gsrc = (const v4f*)(adj + sgl + J);
        v4f t0 = gsrc[0];
        v4f t1 = gsrc[1];
        *(v4f*)&sadj[srow][scol]     = t0;
        *(v4f*)&sadj[srow][scol + 4] = t1;
        __syncthreads();

        // ---- 32 K-steps of fp32 WMMA over this chunk ----
        for (int jj = 0; jj < JCHUNK; jj += 4) {
            const int j0 = J + jj + kb;
            const float ad0 = sadj[m][jj + kb];
            const float ad1 = sadj[m][jj + kb + 1];

            float f0 = esrc + e_dst[(j0)     * N_HEADS + h];
            float f1 = esrc + e_dst[(j0 + 1) * N_HEADS + h];
            f0 = (f0 >= 0.f) ? f0 : NEG_SLOPE * f0;
            f1 = (f1 >= 0.f) ? f1 : NEG_SLOPE * f1;

            v2f a;
            a.x = (ad0 > 0.f) ? __expf(f0 - mi) * inv : 0.f;
            a.y = (ad1 > 0.f) ? __expf(f1 - mi) * inv : 0.f;

            bool nz = (a.x != 0.f) || (a.y != 0.f);
            if (__any(nz)) {                 // wave-uniform -> EXEC stays all-1s
                const float* bp0 = Wh + (size_t)j0 * HID + h * OUT_FEAT + m;
                const float* bp1 = bp0 + HID;
                v2f b0, b1, b2, b3;
                b0.x = bp0[0];  b0.y = bp1[0];
                b1.x = bp0[16]; b1.y = bp1[16];
                b2.x = bp0[32]; b2.y = bp1[32];
                b3.x = bp0[48]; b3.y = bp1[48];
                acc0 = __builtin_amdgcn_wmma_f32_16x16x4_f32(false, a, false, b0,
                                                             (short)0, acc0, false, false);
                acc1 = __builtin_amdgcn_wmma_f32_16x16x4_f32(false, a, false, b1,
                                                             (short)0, acc1, false, false);
                acc2 = __builtin_amdgcn_wmma_f32_16x16x4_f32(false, a, false, b2,
                                                             (short)0, acc2, false, false);
                acc3 = __builtin_amdgcn_wmma_f32_16x16x4_f32(false, a, false, b3,
                                                             (short)0, acc3, false, false);
            }
        }
    }

    // ---- epilogue: relu + store ----
    const int row_hi = (lane >> 4) * 8;
    const int col0   = h * OUT_FEAT + m;
#pragma unroll
    for (int r = 0; r < 8; ++r) {
        const size_t row = (size_t)(i0 + row_hi + r) * HID;
        out[row + col0 +  0] = fmaxf(acc0[r], 0.f);
        out[row + col0 + 16] = fmaxf(acc1[r], 0.f);
        out[row + col0 + 32] = fmaxf(acc2[r], 0.f);
        out[row + col0 + 48] = fmaxf(acc3[r], 0.f);
    }
}

// ---------------------------------------------------------------------------
extern "C" void kernel_launch(void* const* d_in, const int* in_sizes, int n_in,
                              void* d_out, int out_size, void* d_ws, size_t ws_size,
                              hipStream_t stream) {
    const float* x   = (const float*)d_in[0];
    const float* adj = (const float*)d_in[1];
    const float* W   = (const float*)d_in[2];
    const float* a   = (const float*)d_in[3];
    float* out = (float*)d_out;

    // Workspace layout (fp32):  Wh | e_src | e_dst | row_m | row_inv
    float* Wh      = (float*)d_ws;                       // 4096*512 = 2M floats
    float* e_src   = Wh + (size_t)N_NODES * HID;         // 32768
    float* e_dst   = e_src + N_NODES * N_HEADS;          // 32768
    float* row_m   = e_dst + N_NODES * N_HEADS;          // 32768
    float* row_inv = row_m + N_NODES * N_HEADS;          // 32768

    gat_gemm_xw<<<dim3(HID / 64, N_NODES / 16), 128, 0, stream>>>(x, W, Wh);
    gat_scores<<<(N_NODES * N_HEADS) / 256, 256, 0, stream>>>(Wh, a, e_src, e_dst);
    gat_stats<<<N_NODES, 256, 0, stream>>>(adj, e_src, e_dst, row_m, row_inv);
    gat_aggregate<<<N_NODES / 16, 256, 0, stream>>>(adj, Wh, e_src, e_dst,
                                                    row_m, row_inv, out);
}